// SConvLayer_60730837565946
// MI455X (gfx1250) — compile-verified
//
#include <hip/hip_runtime.h>
#include <hip/hip_bf16.h>
#include <math.h>

// ---------------------------------------------------------------------------
// Problem constants: B=4, L=2048, D=512  ->  M = B*L = 8192, N = K = 512
// ---------------------------------------------------------------------------
#define BB   4
#define LL   2048
#define DD   512
#define MM   (BB * LL)          // 8192
#define DECAY 0.99900049983f    // exp(-0.001)

typedef float v2f __attribute__((ext_vector_type(2)));
typedef float v8f __attribute__((ext_vector_type(8)));

// ---------------------------------------------------------------------------
// Kernel 1: GEMM  angle = x @ fc_w^T + fc_b, fused epilogue computing
//   p = exp(-0.001) * (cos(angle*scale) + i sin(angle*scale))   (packed float2)
// Tile: workgroup 128(M) x 128(N), 8 waves, wave tile 32x64 (2x4 WMMA frags).
// K chunked by 16, double-buffered LDS filled with CDNA5 async
// global->LDS copies (global_load_async_to_lds_b128, ASYNCcnt-tracked).
// ---------------------------------------------------------------------------
#define KC         16
#define LDS_STRIDE 20   // 16 + 4 pad floats; 80B row keeps 16B alignment

__device__ __forceinline__ void async_tile16(const float* gptr, const float* lptr)
{
    // CDNA5 async copy: VDST = LDS byte address (wave-relative, low 32 bits of
    // the generic pointer), VADDR = 64-bit global address.  Tracked by ASYNCcnt.
    unsigned lds_off = (unsigned)(unsigned long long)lptr;
    asm volatile("global_load_async_to_lds_b128 %0, %1, off"
                 :: "v"(lds_off), "v"(gptr)
                 : "memory");
}

__global__ __launch_bounds__(256)
void sconv_gemm_phazor(const float* __restrict__ x,
                       const float* __restrict__ fc_w,
                       const float* __restrict__ fc_b,
                       const float* __restrict__ scale,
                       float2* __restrict__ Pp)
{
    __shared__ float As[2][128 * LDS_STRIDE];
    __shared__ float Ws[2][128 * LDS_STRIDE];

    const int tid  = threadIdx.x;
    const int wave = tid >> 5;
    const int lane = tid & 31;
    const int lm   = lane & 15;        // M/N index within a 16-wide frag
    const int kh   = (lane >> 4) << 1; // K pair offset: 0 or 2

    const int wm = wave >> 1;          // 0..3  (M direction, 32 rows each)
    const int wn = wave & 1;           // 0..1  (N direction, 64 cols each)

    const int m_wg = blockIdx.x * 128; // 8192/128 = 64 blocks
    const int n_wg = blockIdx.y * 128; // 512/128  = 4 blocks

    const float* xg = x    + (size_t)m_wg * DD;
    const float* wg = fc_w + (size_t)n_wg * DD;

    // staging slots: 128 rows x 16 floats = 512 float4 per matrix, 2/thread
    const int s_row = (tid + 0)   >> 2;        // rows 0..63   (slot 0)
    const int s_row2 = (tid + 256) >> 2;       // rows 64..127 (slot 1)
    const int s_k4 = (tid & 3) << 2;           // 0,4,8,12

    v8f acc[2][4];
    const v8f vzero = {0.f, 0.f, 0.f, 0.f, 0.f, 0.f, 0.f, 0.f};
#pragma unroll
    for (int mi = 0; mi < 2; ++mi)
#pragma unroll
        for (int ni = 0; ni < 4; ++ni)
            acc[mi][ni] = vzero;

    // prefetch chunk 0 into buffer 0 (4 async b128 copies per thread)
    async_tile16(xg + (size_t)s_row  * DD + s_k4,  &As[0][s_row  * LDS_STRIDE + s_k4]);
    async_tile16(xg + (size_t)s_row2 * DD + s_k4,  &As[0][s_row2 * LDS_STRIDE + s_k4]);
    async_tile16(wg + (size_t)s_row  * DD + s_k4,  &Ws[0][s_row  * LDS_STRIDE + s_k4]);
    async_tile16(wg + (size_t)s_row2 * DD + s_k4,  &Ws[0][s_row2 * LDS_STRIDE + s_k4]);

    const int NCHUNK = DD / KC;   // 32
    for (int c = 0; c < NCHUNK; ++c) {
        const int buf = c & 1;
        if (c + 1 < NCHUNK) {
            // prefetch next chunk into the other buffer (protected by the
            // trailing barrier of the previous iteration's compute phase)
            const int ko = (c + 1) * KC;
            async_tile16(xg + (size_t)s_row  * DD + ko + s_k4, &As[buf ^ 1][s_row  * LDS_STRIDE + s_k4]);
            async_tile16(xg + (size_t)s_row2 * DD + ko + s_k4, &As[buf ^ 1][s_row2 * LDS_STRIDE + s_k4]);
            async_tile16(wg + (size_t)s_row  * DD + ko + s_k4, &Ws[buf ^ 1][s_row  * LDS_STRIDE + s_k4]);
            async_tile16(wg + (size_t)s_row2 * DD + ko + s_k4, &Ws[buf ^ 1][s_row2 * LDS_STRIDE + s_k4]);
            // async copies complete in order: <=4 outstanding means the 4
            // copies of the *current* chunk have landed in LDS
            asm volatile("s_wait_asynccnt 0x4" ::: "memory");
        } else {
            asm volatile("s_wait_asynccnt 0x0" ::: "memory");
        }
        __syncthreads();

        // ---- 4 WMMA K-steps of 4 on the current buffer ----
#pragma unroll
        for (int ks = 0; ks < KC; ks += 4) {
            v2f af[2], bf[4];
#pragma unroll
            for (int mi = 0; mi < 2; ++mi)
                af[mi] = *(const v2f*)(&As[buf][(wm * 32 + mi * 16 + lm) * LDS_STRIDE + ks + kh]);
#pragma unroll
            for (int ni = 0; ni < 4; ++ni)
                bf[ni] = *(const v2f*)(&Ws[buf][(wn * 64 + ni * 16 + lm) * LDS_STRIDE + ks + kh]);
#pragma unroll
            for (int mi = 0; mi < 2; ++mi)
#pragma unroll
                for (int ni = 0; ni < 4; ++ni)
                    acc[mi][ni] = __builtin_amdgcn_wmma_f32_16x16x4_f32(
                        false, af[mi], false, bf[ni],
                        (short)0, acc[mi][ni], false, false);
        }
        __syncthreads();
    }

    // ---- epilogue: bias + phase -> packed p = DECAY*(cos, sin) ----
    const int mhalf = (lane >> 4) << 3;   // 0 or 8 (C-matrix M offset)
#pragma unroll
    for (int mi = 0; mi < 2; ++mi) {
#pragma unroll
        for (int ni = 0; ni < 4; ++ni) {
            const int gn   = n_wg + wn * 64 + ni * 16 + lm;
            const int gm0  = m_wg + wm * 32 + mi * 16 + mhalf;
            const float bias = fc_b[gn];
            const float sc   = scale[gn];
#pragma unroll
            for (int v = 0; v < 8; ++v) {
                float ang = acc[mi][ni][v] + bias;
                float th  = ang * sc;
                float sv, cv;
                __sincosf(th, &sv, &cv);
                size_t o = (size_t)(gm0 + v) * DD + gn;
                Pp[o] = make_float2(DECAY * cv, DECAY * sv);
            }
        }
    }
}

// ---------------------------------------------------------------------------
// Scan: s(t) = p(t)*s(t-1) + x(t), s(-1) = last_conv; h = Re(s)
// Chunked: NC=16 chunks of CH=128 per channel (channel = (b,d), B*D = 2048)
// ---------------------------------------------------------------------------
#define NC 16
#define CH 128

__global__ __launch_bounds__(256)
void sconv_scan_partial(const float* __restrict__ x,
                        const float2* __restrict__ Pp,
                        float* __restrict__ Ar, float* __restrict__ Ai,
                        float* __restrict__ Sr, float* __restrict__ Si)
{
    const int g = blockIdx.x * blockDim.x + threadIdx.x;   // 0..32767
    const int d = g & (DD - 1);
    const int c = (g >> 9) & (NC - 1);
    const int b = g >> 13;
    size_t base = ((size_t)b * LL + (size_t)c * CH) * DD + d;

    float ar = 1.f, ai = 0.f, sr = 0.f, si = 0.f;
    for (int i = 0; i < CH; ++i) {
        size_t o = base + (size_t)i * DD;
        float2 p = Pp[o];
        float xv = x[o];
        float nsr = fmaf(p.x, sr, fmaf(-p.y, si, xv));
        float nsi = fmaf(p.x, si, p.y * sr);
        sr = nsr; si = nsi;
        float nar = fmaf(p.x, ar, -p.y * ai);
        float nai = fmaf(p.x, ai, p.y * ar);
        ar = nar; ai = nai;
    }
    Ar[g] = ar; Ai[g] = ai; Sr[g] = sr; Si[g] = si;
}

__global__ __launch_bounds__(256)
void sconv_scan_combine(const float* __restrict__ lc_re,
                        const float* __restrict__ lc_im,
                        const float* __restrict__ Ar, const float* __restrict__ Ai,
                        const float* __restrict__ Sr, const float* __restrict__ Si,
                        float* __restrict__ Ir, float* __restrict__ Ii)
{
    const int g = blockIdx.x * blockDim.x + threadIdx.x;   // 0..2047
    const int d = g & (DD - 1);
    const int b = g >> 9;
    float sr = lc_re[d], si = lc_im[d];
#pragma unroll
    for (int c = 0; c < NC; ++c) {
        int idx = ((b * NC + c) << 9) + d;
        Ir[idx] = sr; Ii[idx] = si;
        float ar = Ar[idx], ai = Ai[idx];
        float nsr = fmaf(ar, sr, fmaf(-ai, si, Sr[idx]));
        float nsi = fmaf(ar, si, fmaf(ai, sr, Si[idx]));
        sr = nsr; si = nsi;
    }
}

__global__ __launch_bounds__(256)
void sconv_scan_final(const float* __restrict__ x,
                      const float2* __restrict__ Pp,
                      const float* __restrict__ Ir,
                      const float* __restrict__ Ii,
                      float* __restrict__ h)
{
    const int g = blockIdx.x * blockDim.x + threadIdx.x;   // 0..32767
    const int d = g & (DD - 1);
    const int c = (g >> 9) & (NC - 1);
    const int b = g >> 13;
    size_t base = ((size_t)b * LL + (size_t)c * CH) * DD + d;

    float sr = Ir[g], si = Ii[g];
    for (int i = 0; i < CH; ++i) {
        size_t o = base + (size_t)i * DD;
        float2 p = Pp[o];
        float xv = x[o];
        float nsr = fmaf(p.x, sr, fmaf(-p.y, si, xv));
        float nsi = fmaf(p.x, si, p.y * sr);
        sr = nsr; si = nsi;
        h[o] = sr;
    }
}

// ---------------------------------------------------------------------------
// LayerNorm over D (in place on d_out), one 256-thread block per row
// ---------------------------------------------------------------------------
__global__ __launch_bounds__(256)
void sconv_layernorm(float* __restrict__ hy,
                     const float* __restrict__ gamma,
                     const float* __restrict__ beta)
{
    __shared__ float s1[256];
    __shared__ float s2[256];
    const int tid = threadIdx.x;
    const size_t row = (size_t)blockIdx.x * DD;

    float v0 = hy[row + tid];
    float v1 = hy[row + tid + 256];
    s1[tid] = v0 + v1;
    s2[tid] = v0 * v0 + v1 * v1;
    __syncthreads();
#pragma unroll
    for (int s = 128; s > 0; s >>= 1) {
        if (tid < s) { s1[tid] += s1[tid + s]; s2[tid] += s2[tid + s]; }
        __syncthreads();
    }
    const float mu   = s1[0] * (1.0f / DD);
    const float var  = s2[0] * (1.0f / DD) - mu * mu;
    const float rstd = rsqrtf(var + 1e-5f);

    hy[row + tid]       = (v0 - mu) * rstd * gamma[tid]       + beta[tid];
    hy[row + tid + 256] = (v1 - mu) * rstd * gamma[tid + 256] + beta[tid + 256];
}

// ---------------------------------------------------------------------------
// Launcher
// inputs: x, fc_w, fc_b, phazor_angle_scale, last_conv_re, last_conv_im,
//         ln_gamma, ln_beta
// ---------------------------------------------------------------------------
extern "C" void kernel_launch(void* const* d_in, const int* in_sizes, int n_in,
                              void* d_out, int out_size, void* d_ws, size_t ws_size,
                              hipStream_t stream)
{
    const float* x      = (const float*)d_in[0];
    const float* fc_w   = (const float*)d_in[1];
    const float* fc_b   = (const float*)d_in[2];
    const float* scale  = (const float*)d_in[3];
    const float* lc_re  = (const float*)d_in[4];
    const float* lc_im  = (const float*)d_in[5];
    const float* gamma  = (const float*)d_in[6];
    const float* beta   = (const float*)d_in[7];
    float* out = (float*)d_out;

    const size_t NTOT = (size_t)MM * DD;      // 4,194,304 elements
    const size_t NAGG = (size_t)BB * DD * NC; // 32,768

    float2* Pp = (float2*)d_ws;               // packed (p_re, p_im): 32 MB
    float*  Ar = (float*)(Pp + NTOT);
    float*  Ai = Ar + NAGG;
    float*  Sr = Ai + NAGG;
    float*  Si = Sr + NAGG;
    float*  Ir = Si + NAGG;
    float*  Ii = Ir + NAGG;

    // 1) GEMM + phazor epilogue (f32 WMMA, async LDS staging)
    dim3 ggrid(MM / 128, DD / 128);
    sconv_gemm_phazor<<<ggrid, 256, 0, stream>>>(x, fc_w, fc_b, scale, Pp);

    // 2) chunked complex scan
    sconv_scan_partial<<<(BB * DD * NC) / 256, 256, 0, stream>>>(x, Pp, Ar, Ai, Sr, Si);
    sconv_scan_combine<<<(BB * DD) / 256, 256, 0, stream>>>(lc_re, lc_im, Ar, Ai, Sr, Si, Ir, Ii);
    sconv_scan_final<<<(BB * DD * NC) / 256, 256, 0, stream>>>(x, Pp, Ir, Ii, out);

    // 3) LayerNorm over D, in place on d_out
    sconv_layernorm<<<MM, 256, 0, stream>>>(out, gamma, beta);
}